// MembraneIntegrator_18683107737650
// MI455X (gfx1250) — compile-verified
//
#include <hip/hip_runtime.h>
#include <cstdint>

// MembraneIntegrator: m_t = beta*m_{t-1} + (1-beta)*x_t, per-channel beta.
// Memory-bound streaming scan (268 MB @ 23.3 TB/s => ~11.5us floor).
// Pass 0: beta[D] precompute (kills ~2M redundant transcendentals).
// Pass 1: chunk partials (x streams into 192MB L2).
// Pass 2: chunk-carry combine (beta^32 per channel).
// Pass 3: re-scan (x from L2) via async-to-LDS pipeline + NT output stores.

typedef float v4f __attribute__((ext_vector_type(4)));

constexpr int L  = 2048;
constexpr int BS = 32;
constexpr int D  = 512;
constexpr int N  = BS * D;   // 16384 floats per timestep (64 KB row)
constexpr int NV = N / 4;    // 4096 float4 channel groups
constexpr int T  = 32;       // timesteps per chunk
constexpr int C  = L / T;    // 64 chunks

// ws layout: P[C*NV] chunk states (4 MB), then Beta[NV] (64 KB)
constexpr size_t WS_P_ELEMS    = (size_t)C * NV;
constexpr size_t WS_BETA_ELEMS = (size_t)NV;

// ---------- helpers ----------

__device__ __forceinline__ v4f compute_beta(const float* __restrict__ w_tau, int d0) {
  v4f b;
#pragma unroll
  for (int j = 0; j < 4; ++j) {
    float w   = w_tau[d0 + j];
    float sp  = fmaxf(w, 0.0f) + log1pf(expf(-fabsf(w)));  // stable softplus
    float tau = sp + 1.0f;                                  // tau >= 1
    float bb  = 1.0f - 1.0f / tau;
    b[j] = fminf(fmaxf(bb, 0.0f), 0.9999f);
  }
  return b;
}

// CDNA5 async DMA: global -> LDS, 16B per lane, tracked by ASYNCcnt.
__device__ __forceinline__ void async_load_b128(unsigned lds_off, const float* gaddr) {
  asm volatile("global_load_async_to_lds_b128 %0, %1, off"
               :: "v"(lds_off), "v"(gaddr)
               : "memory");
}

// ---------- pass 0: beta precompute (512 channels -> 64 KB, L2-resident) ----------

__global__ void k_beta(const float* __restrict__ w_tau, v4f* __restrict__ Beta) {
  int vc = blockIdx.x * 128 + threadIdx.x;      // [0, NV)
  int d0 = (vc * 4) & (D - 1);
  Beta[vc] = compute_beta(w_tau, d0);
}

// ---------- pass 1: per-chunk local scan (zero init), store chunk-end ----------

__global__ void k_partial(const float* __restrict__ x,
                          const v4f* __restrict__ Beta,
                          v4f* __restrict__ P) {
  int tid   = blockIdx.x * 256 + threadIdx.x;   // [0, C*NV)
  int chunk = tid >> 12;                        // tid / NV
  int vc    = tid & (NV - 1);                   // tid % NV

  v4f beta = Beta[vc];
  v4f omb  = 1.0f - beta;

  const float* xp = x + (size_t)chunk * T * N + (size_t)vc * 4;

  v4f m = {0.f, 0.f, 0.f, 0.f};
#pragma unroll
  for (int t = 0; t < T; ++t) {
    if (t + 8 < T)
      __builtin_prefetch(xp + (size_t)(t + 8) * N, 0, 0);  // global_prefetch_b8
    v4f xv = *(const v4f*)(xp + (size_t)t * N);
    m = beta * m + omb * xv;
  }
  P[(size_t)chunk * NV + vc] = m;   // chunk-local inclusive end (zero-init)
}

// ---------- pass 2: sequential combine over chunks, in-place exclusive ----------

__global__ void k_combine(const v4f* __restrict__ Beta, v4f* __restrict__ P) {
  int vc = blockIdx.x * 128 + threadIdx.x;      // [0, NV)

  v4f beta = Beta[vc];
  v4f b2  = beta * beta;
  v4f b4  = b2 * b2;
  v4f b8  = b4 * b4;
  v4f b16 = b8 * b8;
  v4f bT  = b16 * b16;                          // beta^32

  v4f carry = {0.f, 0.f, 0.f, 0.f};
#pragma unroll 8
  for (int c = 0; c < C; ++c) {
    size_t idx = (size_t)c * NV + vc;
    v4f p = P[idx];
    P[idx] = carry;                             // exclusive start state for chunk c
    carry = bT * carry + p;
  }
}

// ---------- pass 3: re-scan with correct start state; async LDS pipeline ----------

__global__ void k_apply(const float* __restrict__ x,
                        const v4f* __restrict__ Beta,
                        const v4f* __restrict__ P,
                        float* __restrict__ out) {
  __shared__ v4f abuf[4][256];                  // 16 KB, 4-deep per-lane pipeline

  int tid   = blockIdx.x * 256 + threadIdx.x;
  int chunk = tid >> 12;
  int vc    = tid & (NV - 1);

  v4f beta = Beta[vc];
  v4f omb  = 1.0f - beta;

  const float* xp = x   + (size_t)chunk * T * N + (size_t)vc * 4;
  float*       op = out + (size_t)chunk * T * N + (size_t)vc * 4;

  unsigned loff[4];
#pragma unroll
  for (int s = 0; s < 4; ++s)
    loff[s] = (unsigned)(uintptr_t)&abuf[s][threadIdx.x];   // low 32 bits = LDS offset

  // prologue: 3 timesteps in flight
#pragma unroll
  for (int s = 0; s < 3; ++s)
    async_load_b128(loff[s], xp + (size_t)s * N);

  v4f m = P[(size_t)chunk * NV + vc];           // correct carry-in for this chunk

#pragma unroll
  for (int t = 0; t < T; ++t) {
    if (t + 3 < T) {
      async_load_b128(loff[(t + 3) & 3], xp + (size_t)(t + 3) * N);
      asm volatile("s_wait_asynccnt 3" ::: "memory");   // oldest (slot t) complete
    } else {
      asm volatile("s_wait_asynccnt 0" ::: "memory");   // drain tail
    }
    v4f xv = abuf[t & 3][threadIdx.x];          // ds_load_b128
    m = beta * m + omb * xv;
    // NT store: keep the 134MB output stream from evicting x out of the 192MB L2
    __builtin_nontemporal_store(m, (v4f*)(op + (size_t)t * N));
  }
}

// ---------- fallback: single-pass scan (if workspace too small) ----------

__global__ void k_naive(const float* __restrict__ x,
                        const float* __restrict__ w_tau,
                        float* __restrict__ out) {
  int vc = blockIdx.x * 256 + threadIdx.x;      // [0, NV)
  int d0 = (vc * 4) & (D - 1);
  v4f beta = compute_beta(w_tau, d0);
  v4f omb  = 1.0f - beta;

  const float* xp = x   + (size_t)vc * 4;
  float*       op = out + (size_t)vc * 4;
  v4f m = {0.f, 0.f, 0.f, 0.f};
  for (int t = 0; t < L; ++t) {
    v4f xv = *(const v4f*)(xp + (size_t)t * N);
    m = beta * m + omb * xv;
    __builtin_nontemporal_store(m, (v4f*)(op + (size_t)t * N));
  }
}

// ---------- launcher ----------

extern "C" void kernel_launch(void* const* d_in, const int* in_sizes, int n_in,
                              void* d_out, int out_size, void* d_ws, size_t ws_size,
                              hipStream_t stream) {
  const float* x     = (const float*)d_in[0];   // [L, BS, D] f32
  const float* w_tau = (const float*)d_in[1];   // [D] f32
  float*       out   = (float*)d_out;           // [L, BS, D] f32

  const size_t need = (WS_P_ELEMS + WS_BETA_ELEMS) * sizeof(v4f);  // ~4.06 MB

  if (ws_size >= need) {
    v4f* P    = (v4f*)d_ws;
    v4f* Beta = P + WS_P_ELEMS;
    k_beta   <<<NV / 128, 128, 0, stream>>>(w_tau, Beta);
    k_partial<<<(C * NV) / 256, 256, 0, stream>>>(x, Beta, P);
    k_combine<<<NV / 128, 128, 0, stream>>>(Beta, P);
    k_apply  <<<(C * NV) / 256, 256, 0, stream>>>(x, Beta, P, out);
  } else {
    k_naive<<<NV / 256, 256, 0, stream>>>(x, w_tau, out);
  }
}